// ResidualGraphBlock_9990093931292
// MI455X (gfx1250) — compile-verified
//
#include <hip/hip_runtime.h>

#define N_NODES 10000
#define E_EDGES 640000
#define D_DIM   128
#define ALPHA   0.2f
#define LN_EPS  1e-5f

typedef __attribute__((ext_vector_type(2))) float v2f;
typedef __attribute__((ext_vector_type(8))) float v8f;

// ---------------------------------------------------------------------------
// Kernel 0: zero the aggregation buffer (atomics accumulate into it)
// ---------------------------------------------------------------------------
__global__ __launch_bounds__(256) void zero_f4(float4* __restrict__ p, int n4) {
    int i = blockIdx.x * blockDim.x + threadIdx.x;
    if (i < n4) p[i] = make_float4(0.f, 0.f, 0.f, 0.f);
}

// ---------------------------------------------------------------------------
// Kernel 1: h_lin = h @ W^T via V_WMMA_F32_16X16X4_F32
//   block = 256 threads = 8 waves; each wave computes 16 rows x 128 cols.
//   W is staged TRANSPOSED in LDS (Wt[d][kout]) so B-operand reads are
//   bank-conflict-free (lanes 0..15 hit consecutive kout -> consecutive banks).
// ---------------------------------------------------------------------------
__global__ __launch_bounds__(256) void gemm_wmma_f32(const float* __restrict__ h,
                                                     const float* __restrict__ W,
                                                     float* __restrict__ h_lin) {
    __shared__ float Wt[D_DIM * D_DIM];   // Wt[d*128 + kout] = W[kout*128 + d]; 64 KB

    const int tid = threadIdx.x;
    // Cooperative transpose-load of W into LDS. r consecutive across lanes ->
    // LDS store banks are consecutive (conflict-free).
    for (int i = tid; i < D_DIM * 32; i += 256) {
        const int r  = i & 127;   // kout (row of W)
        const int c4 = i >> 7;    // float4 column index 0..31
        const float4 w4 = *(const float4*)(W + r * D_DIM + c4 * 4);
        Wt[(c4 * 4 + 0) * D_DIM + r] = w4.x;
        Wt[(c4 * 4 + 1) * D_DIM + r] = w4.y;
        Wt[(c4 * 4 + 2) * D_DIM + r] = w4.z;
        Wt[(c4 * 4 + 3) * D_DIM + r] = w4.w;
    }
    __syncthreads();

    const int wave = tid >> 5;
    const int lane = tid & 31;
    const int l16  = lane & 15;
    const int hi   = lane >> 4;                    // 0: K,K+1 | 1: K+2,K+3 half
    const int row0 = blockIdx.x * 128 + wave * 16; // 16 rows per wave
    const int arow = row0 + l16;
    const bool arow_ok = (arow < N_NODES);

    v8f acc[8];
#pragma unroll
    for (int nt = 0; nt < 8; ++nt)
        acc[nt] = (v8f){0.f, 0.f, 0.f, 0.f, 0.f, 0.f, 0.f, 0.f};

    for (int kk = 0; kk < 32; ++kk) {
        const int d = kk * 4 + hi * 2;
        v2f a = {0.f, 0.f};
        if (arow_ok) a = *(const v2f*)(h + arow * D_DIM + d);
#pragma unroll
        for (int nt = 0; nt < 8; ++nt) {
            const int kout = nt * 16 + l16;
            v2f b;
            b.x = Wt[d * D_DIM + kout];
            b.y = Wt[(d + 1) * D_DIM + kout];
            // D = A(16x4 f32) * B(4x16 f32) + C(16x16 f32)
            acc[nt] = __builtin_amdgcn_wmma_f32_16x16x4_f32(
                false, a, false, b, (short)0, acc[nt], false, false);
        }
    }

    // C/D layout: VGPR r -> row (row0 + r + 8*hi), col (nt*16 + l16)
#pragma unroll
    for (int r = 0; r < 8; ++r) {
        const int orow = row0 + r + 8 * hi;
        if (orow < N_NODES) {
#pragma unroll
            for (int nt = 0; nt < 8; ++nt)
                h_lin[orow * D_DIM + nt * 16 + l16] = acc[nt][r];
        }
    }
}

// ---------------------------------------------------------------------------
// Kernel 2: per-edge gather/scale/scatter-add. One wave32 per edge; each lane
// covers 4 of the 128 dims. row/col/weight are wave-uniform -> scalar loads.
// agg (5.12 MB) is L2-resident; non-returning global_atomic_add_f32 scatters.
// ---------------------------------------------------------------------------
__global__ __launch_bounds__(256) void edge_scatter(const float* __restrict__ h_lin,
                                                    const float* __restrict__ nw,
                                                    const int*   __restrict__ row,
                                                    const int*   __restrict__ col,
                                                    float*       __restrict__ agg) {
    const int gid  = blockIdx.x * blockDim.x + threadIdx.x;
    const int e    = gid >> 5;
    const int lane = threadIdx.x & 31;
    if (e >= E_EDGES) return;

    const int src = __builtin_amdgcn_readfirstlane(col[e]);
    const int dst = __builtin_amdgcn_readfirstlane(row[e]);
    const float w = nw[e];

    const float4 v = *(const float4*)(h_lin + src * D_DIM + lane * 4);
    float* a = agg + dst * D_DIM + lane * 4;
    atomicAdd(a + 0, v.x * w);
    atomicAdd(a + 1, v.y * w);
    atomicAdd(a + 2, v.z * w);
    atomicAdd(a + 3, v.w * w);
}

// ---------------------------------------------------------------------------
// Kernel 3: LayerNorm + ReLU + scaled residual. One wave32 per node, float4
// per lane, wave-wide reductions via __shfl_xor (wave32).
// ---------------------------------------------------------------------------
__global__ __launch_bounds__(256) void ln_relu_res(const float* __restrict__ agg,
                                                   const float* __restrict__ h0,
                                                   const float* __restrict__ gamma,
                                                   const float* __restrict__ beta,
                                                   float*       __restrict__ out) {
    const int node = blockIdx.x * 8 + (threadIdx.x >> 5);
    const int lane = threadIdx.x & 31;
    if (node >= N_NODES) return;

    const float4 a = *(const float4*)(agg + node * D_DIM + lane * 4);
    float s = a.x + a.y + a.z + a.w;
#pragma unroll
    for (int off = 16; off >= 1; off >>= 1) s += __shfl_xor(s, off, 32);
    const float mu = s * (1.0f / D_DIM);

    const float4 d = make_float4(a.x - mu, a.y - mu, a.z - mu, a.w - mu);
    float ss = d.x * d.x + d.y * d.y + d.z * d.z + d.w * d.w;
#pragma unroll
    for (int off = 16; off >= 1; off >>= 1) ss += __shfl_xor(ss, off, 32);
    const float inv = rsqrtf(ss * (1.0f / D_DIM) + LN_EPS);

    const float4 g   = *(const float4*)(gamma + lane * 4);
    const float4 b   = *(const float4*)(beta  + lane * 4);
    const float4 h0v = *(const float4*)(h0 + node * D_DIM + lane * 4);

    float4 o;
    o.x = (1.0f - ALPHA) * fmaxf(d.x * inv * g.x + b.x, 0.f) + ALPHA * h0v.x;
    o.y = (1.0f - ALPHA) * fmaxf(d.y * inv * g.y + b.y, 0.f) + ALPHA * h0v.y;
    o.z = (1.0f - ALPHA) * fmaxf(d.z * inv * g.z + b.z, 0.f) + ALPHA * h0v.z;
    o.w = (1.0f - ALPHA) * fmaxf(d.w * inv * g.w + b.w, 0.f) + ALPHA * h0v.w;
    *(float4*)(out + node * D_DIM + lane * 4) = o;
}

// ---------------------------------------------------------------------------
extern "C" void kernel_launch(void* const* d_in, const int* in_sizes, int n_in,
                              void* d_out, int out_size, void* d_ws, size_t ws_size,
                              hipStream_t stream) {
    const float* h   = (const float*)d_in[0];
    const float* h0  = (const float*)d_in[1];
    const float* nw  = (const float*)d_in[2];
    const float* W   = (const float*)d_in[3];
    const float* gam = (const float*)d_in[4];
    const float* bet = (const float*)d_in[5];
    const int*   row = (const int*)d_in[6];
    const int*   col = (const int*)d_in[7];
    float* out = (float*)d_out;

    float* h_lin = (float*)d_ws;                    // N*D floats (5.12 MB)
    float* agg   = h_lin + N_NODES * D_DIM;         // N*D floats (5.12 MB)

    const int n4 = N_NODES * D_DIM / 4;
    zero_f4<<<(n4 + 255) / 256, 256, 0, stream>>>((float4*)agg, n4);

    gemm_wmma_f32<<<(N_NODES + 127) / 128, 256, 0, stream>>>(h, W, h_lin);

    edge_scatter<<<E_EDGES / 8, 256, 0, stream>>>(h_lin, nw, row, col, agg);

    ln_relu_res<<<(N_NODES + 7) / 8, 256, 0, stream>>>(agg, h0, gam, bet, out);
}